// STCAGCN_10118942949645
// MI455X (gfx1250) — compile-verified
//
#include <hip/hip_runtime.h>

typedef float v2f __attribute__((ext_vector_type(2)));
typedef float v8f __attribute__((ext_vector_type(8)));

#define NNODES 325
#define KPAD   328   // K dimension padded to multiple of 4 (zero-filled)
#define NPAD   328   // row stride of big buffers per batch (3 zero pad rows)
#define TSTEPS 288
#define HIDDIM 64

#define WMMA4(ACC, AF, BF) \
  ACC = __builtin_amdgcn_wmma_f32_16x16x4_f32(false, AF, false, BF, (short)0, ACC, false, false)

// ---------------------------------------------------------------------------
// Diffusion GEMM: Out[s][b] = A_s[b] @ X[b],  A: 325x328(padded), X: 328 x ncols
// (pad rows zero).  Wave = 32x64 tile (8 accumulators, 8 WMMA / K-step).
// 2-deep software pipeline, 32-bit saddr+voffset addressing.
// ---------------------------------------------------------------------------
__global__ __launch_bounds__(256) void k_diffuse(
    const float* __restrict__ Af, const float* __restrict__ Ab,
    const float* __restrict__ Aaa, const float* __restrict__ X,
    float* __restrict__ Dfa, float* __restrict__ Dba, float* __restrict__ Daa,
    int ncols)
{
  int s = blockIdx.z % 3, b = blockIdx.z / 3;
  const float* A = (s == 0) ? Af : (s == 1) ? Ab : (Aaa + (size_t)b * NNODES * KPAD);
  float* Out = ((s == 0) ? Dfa : (s == 1) ? Dba : Daa) + (size_t)b * NPAD * ncols;
  const float* Xb = X + (size_t)b * NPAD * ncols;

  int lane = threadIdx.x & 31, wave = threadIdx.x >> 5;
  int l15 = lane & 15, kh = lane >> 4;
  int m0 = blockIdx.y * 128 + (wave >> 1) * 32;   // 4 M-waves
  int n0 = blockIdx.x * 128 + (wave & 1) * 64;    // 2 N-waves

  // Clamped indices: out-of-range rows/cols only affect outputs that the
  // guarded epilogue never stores, so in-bounds garbage is harmless.
  int ar0 = m0 + l15;      if (ar0 >= NNODES) ar0 = NNODES - 1;
  int ar1 = m0 + 16 + l15; if (ar1 >= NNODES) ar1 = NNODES - 1;
  int nc = n0 + l15;
  int c0 = (nc      < ncols) ? nc      : 0;
  int c1 = (nc + 16 < ncols) ? nc + 16 : 0;
  int c2 = (nc + 32 < ncols) ? nc + 32 : 0;
  int c3 = (nc + 48 < ncols) ? nc + 48 : 0;

  int aidx0 = ar0 * KPAD + 2 * kh;   // 32-bit element indices (saddr+voffset)
  int aidx1 = ar1 * KPAD + 2 * kh;
  int xk0   = (2 * kh) * ncols;

  v8f aA0 = {}, aA1 = {}, aA2 = {}, aA3 = {};
  v8f aB0 = {}, aB1 = {}, aB2 = {}, aB3 = {};

  v2f a0A, a1A, b0A, b1A, b2A, b3A;
  v2f a0B, a1B, b0B, b1B, b2B, b3B;

#define LOADD(a0v, a1v, b0v, b1v, b2v, b3v, IT)                          \
  {                                                                      \
    int ao = (IT) * 4;                                                   \
    int xo = xk0 + (IT) * (4 * ncols);                                   \
    a0v = *(const v2f*)(A + aidx0 + ao);                                 \
    a1v = *(const v2f*)(A + aidx1 + ao);                                 \
    b0v.x = Xb[xo + c0]; b0v.y = Xb[xo + ncols + c0];                    \
    b1v.x = Xb[xo + c1]; b1v.y = Xb[xo + ncols + c1];                    \
    b2v.x = Xb[xo + c2]; b2v.y = Xb[xo + ncols + c2];                    \
    b3v.x = Xb[xo + c3]; b3v.y = Xb[xo + ncols + c3];                    \
  }
#define WMMA8(a0v, a1v, b0v, b1v, b2v, b3v)                              \
  WMMA4(aA0, a0v, b0v); WMMA4(aA1, a0v, b1v);                            \
  WMMA4(aA2, a0v, b2v); WMMA4(aA3, a0v, b3v);                            \
  WMMA4(aB0, a1v, b0v); WMMA4(aB1, a1v, b1v);                            \
  WMMA4(aB2, a1v, b2v); WMMA4(aB3, a1v, b3v);

  LOADD(a0A, a1A, b0A, b1A, b2A, b3A, 0)
  LOADD(a0B, a1B, b0B, b1B, b2B, b3B, 1)
  for (int it = 0; it + 2 < KPAD / 4; it += 2) {     // KPAD/4 = 82 (even)
    WMMA8(a0A, a1A, b0A, b1A, b2A, b3A)
    LOADD(a0A, a1A, b0A, b1A, b2A, b3A, it + 2)
    WMMA8(a0B, a1B, b0B, b1B, b2B, b3B)
    LOADD(a0B, a1B, b0B, b1B, b2B, b3B, it + 3)
  }
  WMMA8(a0A, a1A, b0A, b1A, b2A, b3A)
  WMMA8(a0B, a1B, b0B, b1B, b2B, b3B)
#undef LOADD

#pragma unroll
  for (int r = 0; r < 8; ++r) {
    int row0 = m0 + r + 8 * kh;
    if (row0 < NNODES) {
      int o = row0 * ncols;
      if (nc      < ncols) Out[o + nc     ] = aA0[r];
      if (nc + 16 < ncols) Out[o + nc + 16] = aA1[r];
      if (nc + 32 < ncols) Out[o + nc + 32] = aA2[r];
      if (nc + 48 < ncols) Out[o + nc + 48] = aA3[r];
    }
    int row1 = row0 + 16;
    if (row1 < NNODES) {
      int o = row1 * ncols;
      if (nc      < ncols) Out[o + nc     ] = aB0[r];
      if (nc + 16 < ncols) Out[o + nc + 16] = aB1[r];
      if (nc + 32 < ncols) Out[o + nc + 32] = aB2[r];
      if (nc + 48 < ncols) Out[o + nc + 48] = aB3[r];
    }
  }
}

// ---------------------------------------------------------------------------
// Fused projection(+bias)x3 + concat + fusion(+bias) + ReLU folded into one
// GEMM: Out[t,h] = relu( sum_{s,f} Src_s[f*T+t] * Wc[s*F+f,h] + cb[h] ).
// Per (b,n): (288 x 4F) @ (4F x 64).  Wave = 32x64 tile, 2-deep pipeline.
// ---------------------------------------------------------------------------
__global__ __launch_bounds__(128) void k_projfuse(
    const float* __restrict__ Xin, const float* __restrict__ Dfa,
    const float* __restrict__ Dba, const float* __restrict__ Daa,
    const float* __restrict__ Wc, const float* __restrict__ cb,
    float* __restrict__ Hout, int F, int Fshift)
{
  int bn = blockIdx.y;
  int b = bn / NNODES, n = bn % NNODES;
  int pbn = b * NPAD + n;
  int wave = threadIdx.x >> 5, lane = threadIdx.x & 31;
  int l15 = lane & 15, kh = lane >> 4;
  int mt = blockIdx.x * 4 + wave;
  if (mt >= TSTEPS / 32) return;   // wave-uniform: EXEC stays all-ones
  int t0 = mt * 32;
  int tr0 = t0 + l15;
  int tr1 = t0 + 16 + l15;

  size_t soff = (size_t)pbn * F * TSTEPS;
  const float* src[4] = {Xin + soff, Dfa + soff, Dba + soff, Daa + soff};
  int fmask = F - 1;

  v8f aA0 = {}, aA1 = {}, aA2 = {}, aA3 = {};
  v8f aB0 = {}, aB1 = {}, aB2 = {}, aB3 = {};

  v2f a0A, a1A, b0A, b1A, b2A, b3A;
  v2f a0B, a1B, b0B, b1B, b2B, b3B;

#define LOADP(a0v, a1v, b0v, b1v, b2v, b3v, IT)                          \
  {                                                                      \
    int kn = (IT) * 4 + 2 * kh;                                          \
    int ps0 = kn >> Fshift,       pf0 = (kn & fmask) * TSTEPS;           \
    int ps1 = (kn + 1) >> Fshift, pf1 = ((kn + 1) & fmask) * TSTEPS;     \
    a0v.x = src[ps0][pf0 + tr0]; a0v.y = src[ps1][pf1 + tr0];            \
    a1v.x = src[ps0][pf0 + tr1]; a1v.y = src[ps1][pf1 + tr1];            \
    int wo = kn * 64 + l15;                                              \
    b0v.x = Wc[wo     ]; b0v.y = Wc[wo + 64 ];                           \
    b1v.x = Wc[wo + 16]; b1v.y = Wc[wo + 80 ];                           \
    b2v.x = Wc[wo + 32]; b2v.y = Wc[wo + 96 ];                           \
    b3v.x = Wc[wo + 48]; b3v.y = Wc[wo + 112];                           \
  }

  LOADP(a0A, a1A, b0A, b1A, b2A, b3A, 0)
  if (F > 1) {                                   // kIters = F (even when 64)
    LOADP(a0B, a1B, b0B, b1B, b2B, b3B, 1)
    for (int it = 0; it + 2 < F; it += 2) {
      WMMA8(a0A, a1A, b0A, b1A, b2A, b3A)
      LOADP(a0A, a1A, b0A, b1A, b2A, b3A, it + 2)
      WMMA8(a0B, a1B, b0B, b1B, b2B, b3B)
      LOADP(a0B, a1B, b0B, b1B, b2B, b3B, it + 3)
    }
    WMMA8(a0A, a1A, b0A, b1A, b2A, b3A)
    WMMA8(a0B, a1B, b0B, b1B, b2B, b3B)
  } else {
    WMMA8(a0A, a1A, b0A, b1A, b2A, b3A)
  }
#undef LOADP
#undef WMMA8

  float* Ho = Hout + (size_t)pbn * HIDDIM * TSTEPS;
  float cb0 = cb[l15], cb1 = cb[l15 + 16], cb2 = cb[l15 + 32], cb3 = cb[l15 + 48];
#pragma unroll
  for (int r = 0; r < 8; ++r) {
    int ta = t0 + r + 8 * kh;
    int tb = ta + 16;
    Ho[(l15     ) * TSTEPS + ta] = fmaxf(aA0[r] + cb0, 0.f);
    Ho[(l15 + 16) * TSTEPS + ta] = fmaxf(aA1[r] + cb1, 0.f);
    Ho[(l15 + 32) * TSTEPS + ta] = fmaxf(aA2[r] + cb2, 0.f);
    Ho[(l15 + 48) * TSTEPS + ta] = fmaxf(aA3[r] + cb3, 0.f);
    Ho[(l15     ) * TSTEPS + tb] = fmaxf(aB0[r] + cb0, 0.f);
    Ho[(l15 + 16) * TSTEPS + tb] = fmaxf(aB1[r] + cb1, 0.f);
    Ho[(l15 + 32) * TSTEPS + tb] = fmaxf(aB2[r] + cb2, 0.f);
    Ho[(l15 + 48) * TSTEPS + tb] = fmaxf(aB3[r] + cb3, 0.f);
  }
}

// ---------------------------------------------------------------------------
// Combined-weight precompute (exact linear-algebra fold of proj+fusion).
// ---------------------------------------------------------------------------
__global__ void k_wcomb(const float* __restrict__ ts, const float* __restrict__ bs,
                        const float* __restrict__ ta, const float* __restrict__ ba,
                        const float* __restrict__ Wf, const float* __restrict__ fb,
                        float* __restrict__ Wc, float* __restrict__ cb,
                        int F, int Fshift)
{
  int k = blockIdx.x;          // 0 .. 4F-1
  int h = threadIdx.x;         // 0 .. 63
  int s = k >> Fshift;
  int f = k & (F - 1);
  const float* w0 = Wf + h * 192;
  float acc = 0.f;
  if (s == 0) {
    const float* r1 = ts + (3 * f) * 64;
    const float* r2 = ta + (3 * f) * 64;
    for (int o = 0; o < 64; ++o)
      acc += r1[o] * (w0[o] + w0[64 + o]) + r2[o] * w0[128 + o];
  } else if (s == 1) {
    const float* r1 = ts + (3 * f + 1) * 64;
    for (int o = 0; o < 64; ++o) acc += r1[o] * w0[o];
  } else if (s == 2) {
    const float* r1 = ts + (3 * f + 2) * 64;
    for (int o = 0; o < 64; ++o) acc += r1[o] * w0[o];
  } else {
    const float* r1 = ts + (3 * f + 1) * 64;
    const float* r2 = ts + (3 * f + 2) * 64;
    const float* q1 = ta + (3 * f + 1) * 64;
    const float* q2 = ta + (3 * f + 2) * 64;
    for (int o = 0; o < 64; ++o)
      acc += (r1[o] + r2[o]) * w0[64 + o] + (q1[o] + q2[o]) * w0[128 + o];
  }
  Wc[k * 64 + h] = acc;
  if (k == 0) {
    float c = fb[h];
    for (int o = 0; o < 64; ++o)
      c += bs[o] * (w0[o] + w0[64 + o]) + ba[o] * w0[128 + o];
    cb[h] = c;
  }
}

// ---------------------------------------------------------------------------
// PAM chain (tiny): f1 = lrelu(speed^T @ w1^T), f2 = lrelu(f1 @ w2^T),
// norms, thresholded cosine adjacency written in K-padded layout.
// ---------------------------------------------------------------------------
__global__ void k_pam1(const float* __restrict__ speed, const float* __restrict__ w1,
                       float* __restrict__ f1)
{
  int bn = blockIdx.x, b = bn / NNODES, n = bn % NNODES;
  int h = threadIdx.x;
  float acc = 0.f;
  for (int t = 0; t < TSTEPS; ++t)
    acc += speed[((size_t)b * TSTEPS + t) * NNODES + n] * w1[h * TSTEPS + t];
  f1[(size_t)bn * 64 + h] = (acc > 0.f) ? acc : 0.2f * acc;
}

__global__ void k_pam2(const float* __restrict__ f1, const float* __restrict__ w2,
                       float* __restrict__ f2, float* __restrict__ norms)
{
  __shared__ float sf[64];
  __shared__ float red[64];
  int bn = blockIdx.x, h = threadIdx.x;
  sf[h] = f1[(size_t)bn * 64 + h];
  __syncthreads();
  float acc = 0.f;
#pragma unroll
  for (int j = 0; j < 64; ++j) acc += sf[j] * w2[h * 64 + j];
  acc = (acc > 0.f) ? acc : 0.2f * acc;
  f2[(size_t)bn * 64 + h] = acc;
  red[h] = acc * acc;
  __syncthreads();
  for (int s = 32; s > 0; s >>= 1) {
    if (h < s) red[h] += red[h + s];
    __syncthreads();
  }
  if (h == 0) norms[bn] = fmaxf(sqrtf(red[0]), 1e-8f);
}

__global__ void k_pam3(const float* __restrict__ f2, const float* __restrict__ norms,
                       float* __restrict__ aa)
{
  int b = blockIdx.x / NNODES, i = blockIdx.x % NNODES;
  const float* fi = f2 + ((size_t)b * NNODES + i) * 64;
  float ni = norms[b * NNODES + i];
  float* row = aa + ((size_t)b * NNODES + i) * KPAD;
  for (int j = threadIdx.x; j < KPAD; j += blockDim.x) {
    float outv = 0.f;
    if (j < NNODES) {
      const float* fj = f2 + ((size_t)b * NNODES + j) * 64;
      float d = 0.f;
#pragma unroll
      for (int h = 0; h < 64; ++h) d += fi[h] * fj[h];
      float v = d / (ni * norms[b * NNODES + j]);
      outv = (v > 0.1f) ? v : 0.f;
    }
    row[j] = outv;
  }
}

// Pad a 325x325 adjacency to 325x328 (zero K-pad columns).
__global__ void k_padA(const float* __restrict__ Ain, float* __restrict__ Aout)
{
  int i = blockIdx.x;
  for (int j = threadIdx.x; j < KPAD; j += blockDim.x)
    Aout[(size_t)i * KPAD + j] = (j < NNODES) ? Ain[(size_t)i * NNODES + j] : 0.f;
}

__global__ void k_zero(float* __restrict__ p, size_t nfloats)
{
  size_t i = (size_t)blockIdx.x * blockDim.x + threadIdx.x;
  size_t stride = (size_t)gridDim.x * blockDim.x;
  for (; i < nfloats; i += stride) p[i] = 0.f;
}

// ---------------------------------------------------------------------------
// Temporal conv (kernel 3 along t) + bias + residual, x0 layout.
// ---------------------------------------------------------------------------
__global__ __launch_bounds__(256) void k_temporal(
    const float* __restrict__ Hin, const float* __restrict__ tw,
    const float* __restrict__ tb, float* __restrict__ Hout)
{
  __shared__ float sW[64 * 193];
  __shared__ float sIn[64 * 34];
  int bn = blockIdx.x, b = bn / NNODES, n = bn % NNODES;
  int pbn = b * NPAD + n;
  int t0 = blockIdx.y * 32;
  const float* in = Hin + (size_t)pbn * HIDDIM * TSTEPS;
  for (int i = threadIdx.x; i < 64 * 192; i += 256) {
    int hh = i / 192, r = i % 192;
    sW[hh * 193 + r] = tw[i];
  }
  for (int i = threadIdx.x; i < 64 * 34; i += 256) {
    int ch = i / 34, lt = i % 34;
    int g = t0 - 1 + lt;
    sIn[i] = (g >= 0 && g < TSTEPS) ? in[ch * TSTEPS + g] : 0.f;
  }
  __syncthreads();
  int h = threadIdx.x & 63;
  int tbase = threadIdx.x >> 6;
  float bias = tb[h];
  const float* wrow = &sW[h * 193];
  float* out = Hout + (size_t)pbn * HIDDIM * TSTEPS;
  for (int tt = tbase; tt < 32; tt += 4) {
    float accv = bias + sIn[h * 34 + tt + 1];   // residual
#pragma unroll 8
    for (int i2 = 0; i2 < 64; ++i2) {
      float x0 = sIn[i2 * 34 + tt];
      float x1 = sIn[i2 * 34 + tt + 1];
      float x2 = sIn[i2 * 34 + tt + 2];
      accv += wrow[i2 * 3 + 0] * x0 + wrow[i2 * 3 + 1] * x1 + wrow[i2 * 3 + 2] * x2;
    }
    out[h * TSTEPS + t0 + tt] = accv;
  }
}

// ---------------------------------------------------------------------------
// Streaming readout accumulation into the final output.
// ---------------------------------------------------------------------------
__global__ void k_readout(const float* __restrict__ H, const float* __restrict__ rw,
                          const float* __restrict__ rb, float* __restrict__ out,
                          int layer)
{
  int bn = blockIdx.x, b = bn / NNODES, n = bn % NNODES;
  int pbn = b * NPAD + n;
  int t = threadIdx.x;
  const float* hp = H + (size_t)pbn * HIDDIM * TSTEPS;
  float acc = 0.f;
#pragma unroll 8
  for (int h = 0; h < 64; ++h) acc += hp[h * TSTEPS + t] * rw[layer * 64 + h];
  size_t o = ((size_t)b * TSTEPS + t) * NNODES + n;
  if (layer == 0) out[o] = rb[0] + acc;
  else            out[o] += acc;
}

__global__ void k_transpose_x(const float* __restrict__ x, float* __restrict__ X0)
{
  int bn = blockIdx.x, b = bn / NNODES, n = bn % NNODES;
  int pbn = b * NPAD + n;
  int t = threadIdx.x;
  X0[(size_t)pbn * TSTEPS + t] = x[((size_t)b * TSTEPS + t) * NNODES + n];
}

// ---------------------------------------------------------------------------
extern "C" void kernel_launch(void* const* d_in, const int* in_sizes, int n_in,
                              void* d_out, int out_size, void* d_ws, size_t ws_size,
                              hipStream_t stream)
{
  (void)in_sizes; (void)n_in; (void)out_size; (void)ws_size;
  const float* x       = (const float*)d_in[0];
  const float* fadj    = (const float*)d_in[1];
  const float* badj    = (const float*)d_in[2];
  const float* ffadj   = (const float*)d_in[3];
  const float* fbadj   = (const float*)d_in[4];
  const float* speed   = (const float*)d_in[5];
  const float* pam_w1  = (const float*)d_in[36];
  const float* pam_w2  = (const float*)d_in[37];
  const float* tconv_w = (const float*)d_in[38];
  const float* tconv_b = (const float*)d_in[39];
  const float* rw      = (const float*)d_in[40];
  const float* rb      = (const float*)d_in[41];
  float* out = (float*)d_out;

  const size_t BIGP = (size_t)4 * NPAD * HIDDIM * TSTEPS;   // 24,182,784 floats
  float* ws     = (float*)d_ws;
  float* bufA   = ws;
  float* bufB   = ws + BIGP;
  float* Dfa    = ws + 2 * BIGP;
  float* Dba    = ws + 3 * BIGP;
  float* Daa    = ws + 4 * BIGP;
  float* aa_p   = ws + 5 * BIGP;                           // 4*325*328
  float* fa_p   = aa_p + (size_t)4 * NNODES * KPAD;
  float* ba_p   = fa_p + (size_t)NNODES * KPAD;
  float* ffa_p  = ba_p + (size_t)NNODES * KPAD;
  float* fba_p  = ffa_p + (size_t)NNODES * KPAD;
  float* f1     = fba_p + (size_t)NNODES * KPAD;
  float* f2     = f1 + (size_t)4 * NNODES * 64;
  float* norms  = f2 + (size_t)4 * NNODES * 64;
  float* Wc     = norms + (size_t)4 * NNODES;
  float* cb     = Wc + 256 * 64;

  // Zero bufA+bufB (contiguous): guarantees zero pad rows for every layer view.
  k_zero<<<4096, 256, 0, stream>>>(ws, 2 * BIGP);
  // Pad static adjacencies; PAM writes its adjacency pre-padded.
  k_padA<<<NNODES, 128, 0, stream>>>(fadj, fa_p);
  k_padA<<<NNODES, 128, 0, stream>>>(badj, ba_p);
  k_padA<<<NNODES, 128, 0, stream>>>(ffadj, ffa_p);
  k_padA<<<NNODES, 128, 0, stream>>>(fbadj, fba_p);

  k_transpose_x<<<4 * NNODES, TSTEPS, 0, stream>>>(x, bufA);
  k_pam1<<<4 * NNODES, 64, 0, stream>>>(speed, pam_w1, f1);
  k_pam2<<<4 * NNODES, 64, 0, stream>>>(f1, pam_w2, f2, norms);
  k_pam3<<<4 * NNODES, 128, 0, stream>>>(f2, norms, aa_p);

  float* X = bufA;
  float* H = bufB;
  for (int layer = 0; layer < 5; ++layer) {
    int F = (layer == 0) ? 1 : HIDDIM;
    int Fshift = (layer == 0) ? 0 : 6;
    int ncols = F * TSTEPS;
    const float* Af = (layer == 0) ? ffa_p : fa_p;
    const float* Ab = (layer == 0) ? fba_p : ba_p;
    const float* ts = (const float*)d_in[6 + 6 * layer + 0];
    const float* bs = (const float*)d_in[6 + 6 * layer + 1];
    const float* ta = (const float*)d_in[6 + 6 * layer + 2];
    const float* ba = (const float*)d_in[6 + 6 * layer + 3];
    const float* Wf = (const float*)d_in[6 + 6 * layer + 4];
    const float* fb = (const float*)d_in[6 + 6 * layer + 5];

    // Block tile: 128(M) x 128(N) from 8 waves of 32x64.
    dim3 dgrid((ncols + 127) / 128, (NNODES + 127) / 128, 12);
    k_diffuse<<<dgrid, 256, 0, stream>>>(Af, Ab, aa_p, X, Dfa, Dba, Daa, ncols);

    k_wcomb<<<4 * F, 64, 0, stream>>>(ts, bs, ta, ba, Wf, fb, Wc, cb, F, Fshift);

    // 9 M-tiles of 32 rows per (b,n); 4 waves per block.
    dim3 pgrid((TSTEPS / 32 + 3) / 4, 4 * NNODES);
    k_projfuse<<<pgrid, 128, 0, stream>>>(X, Dfa, Dba, Daa, Wc, cb, H, F, Fshift);

    k_readout<<<4 * NNODES, TSTEPS, 0, stream>>>(H, rw, rb, out, layer);

    if (layer == 0) {
      k_temporal<<<dim3(4 * NNODES, TSTEPS / 32), 256, 0, stream>>>(H, tconv_w, tconv_b, bufA);
      X = bufA; H = bufB;
    } else {
      float* tmp = X; X = H; H = tmp;
    }
  }
}